// EdgeLoss_3547642987419
// MI455X (gfx1250) — compile-verified
//
#include <hip/hip_runtime.h>
#include <cstdint>

// ---------------- problem constants ----------------
constexpr int W = 512, H = 512;
constexpr int TX = 128, TY = 32;            // output tile per workgroup
constexpr int LDSW = 136;                   // TX + 8 (4-float halo each side, float4 aligned)
constexpr int LDSH = 34;                    // TY + 2
constexpr int C4ROW = LDSW / 4;             // 34 float4 chunks per LDS row
constexpr int NCHUNK = LDSH * C4ROW;        // 1156 chunks per image tile
constexpr int TILES_X = W / TX;             // 4
constexpr int TILES_Y = H / TY;             // 16
constexpr int TILES_PP = TILES_X * TILES_Y; // 64 tiles per plane
constexpr int NPLANES = 16 * 3;             // 48 (batch * channels)
constexpr int NBLOCKS = NPLANES * TILES_PP; // 3072 workgroups
constexpr int NTHREADS = 256;               // 8 waves (wave32)
constexpr float EPS = 1e-6f;

typedef __attribute__((ext_vector_type(2))) float v2f;
typedef __attribute__((ext_vector_type(8))) float v8f;

// ---------------- CDNA5 async global -> LDS (16B per lane) ----------------
__device__ __forceinline__ void async_copy16(const float* g, unsigned ldsByteOff) {
    // GLOBAL_LOAD_ASYNC_TO_LDS_B128: VDST = LDS byte offset VGPR, VADDR = 64-bit addr
    asm volatile("global_load_async_to_lds_b128 %0, %1, off"
                 :: "v"(ldsByteOff), "v"(g)
                 : "memory");
}
__device__ __forceinline__ void async_wait0() {
    asm volatile("s_wait_asynccnt 0x0" ::: "memory");
}

// ---------------- 32-lane wave sum via one V_WMMA_F32_16X16X4_F32 ----------------
// A[m][0] = p(lane m), A[m][2] = p(lane m+16), A[m][1]=A[m][3]=0 ; B = ones
// => C[m][n] = p_m + p_{m+16}. Lane 0 holds rows 0..7 of col 0, lane 16 rows 8..15.
__device__ __forceinline__ float wave_sum32(float p) {
#if __has_builtin(__builtin_amdgcn_wmma_f32_16x16x4_f32)
    v2f a; a.x = p;    a.y = 0.0f;
    v2f b; b.x = 1.0f; b.y = 1.0f;
    v8f c = {};
    c = __builtin_amdgcn_wmma_f32_16x16x4_f32(false, a, false, b,
                                              (short)0, c, false, false);
    float t = c[0] + c[1] + c[2] + c[3] + c[4] + c[5] + c[6] + c[7];
    return __shfl(t, 0, 32) + __shfl(t, 16, 32);
#else
    for (int off = 16; off > 0; off >>= 1) p += __shfl_down(p, off, 32);
    return __shfl(p, 0, 32);
#endif
}

// separable Sobel row terms: hx = right-left, hy = [1,2,1] smoothing
__device__ __forceinline__ void rowHV(const float* t, int row, int tx,
                                      float& hx, float& hy) {
    const float c0 = t[row * LDSW + tx + 3];
    const float c1 = t[row * LDSW + tx + 4];
    const float c2 = t[row * LDSW + tx + 5];
    hx = c2 - c0;
    hy = c0 + 2.0f * c1 + c2;
}

__global__ void __launch_bounds__(NTHREADS)
edge_tile_kernel(const float* __restrict__ img1, const float* __restrict__ img2,
                 float* __restrict__ partials) {
    __shared__ float t1[LDSH * LDSW];
    __shared__ float t2[LDSH * LDSW];
    __shared__ float red[NTHREADS];

    const int tid   = threadIdx.x;
    const int bid   = blockIdx.x;
    const int plane = bid / TILES_PP;
    const int tile  = bid % TILES_PP;
    const int tx0   = (tile % TILES_X) * TX;
    const int ty0   = (tile / TILES_X) * TY;
    const long planeOff = (long)plane * (W * H);

    // 1) zero both LDS tiles -> implements zero padding at image borders
    for (int i = tid; i < LDSH * LDSW; i += NTHREADS) { t1[i] = 0.0f; t2[i] = 0.0f; }
    __syncthreads();

    // 2) async-gather both image tiles (incl. halo) into LDS, 16B per lane
    const unsigned l1 = (unsigned)(uintptr_t)(&t1[0]); // low 32 bits of flat LDS addr = LDS offset
    const unsigned l2 = (unsigned)(uintptr_t)(&t2[0]);
    for (int i = tid; i < NCHUNK; i += NTHREADS) {
        const int row = i / C4ROW;
        const int c4  = i - row * C4ROW;
        const int gy  = ty0 - 1 + row;
        const int gx  = tx0 - 4 + c4 * 4;     // 16B-aligned global column
        if ((unsigned)gy < (unsigned)H && (unsigned)gx < (unsigned)W) {
            const long goff   = planeOff + (long)gy * W + gx;
            const unsigned lo = (unsigned)((row * LDSW + c4 * 4) * 4);
            async_copy16(img1 + goff, l1 + lo);
            async_copy16(img2 + goff, l2 + lo);
        }
    }
    async_wait0();
    __syncthreads();

    // 3) separable Sobel with rolling row registers: 16 rows per thread
    const int tx = tid & (TX - 1);
    const int r0 = (tid >> 7) * (TY / 2);     // 0 or 16
    float ax0, ay0, ax1, ay1, bx0, by0, bx1, by1;
    rowHV(t1, r0,     tx, ax0, ay0);
    rowHV(t1, r0 + 1, tx, ax1, ay1);
    rowHV(t2, r0,     tx, bx0, by0);
    rowHV(t2, r0 + 1, tx, bx1, by1);

    float acc = 0.0f;
#pragma unroll
    for (int j = 0; j < TY / 2; ++j) {
        float ax2, ay2, bx2, by2;
        rowHV(t1, r0 + 2 + j, tx, ax2, ay2);
        rowHV(t2, r0 + 2 + j, tx, bx2, by2);
        const float ex1 = ax0 + 2.0f * ax1 + ax2;
        const float ey1 = ay2 - ay0;
        const float ex2 = bx0 + 2.0f * bx1 + bx2;
        const float ey2 = by2 - by0;
        const float m1 = __builtin_sqrtf(__builtin_fmaf(ex1, ex1, __builtin_fmaf(ey1, ey1, EPS)));
        const float m2 = __builtin_sqrtf(__builtin_fmaf(ex2, ex2, __builtin_fmaf(ey2, ey2, EPS)));
        acc += __builtin_fabsf(m1 - m2);
        ax0 = ax1; ay0 = ay1; ax1 = ax2; ay1 = ay2;
        bx0 = bx1; by0 = by1; bx1 = bx2; by1 = by2;
    }

    // 4) block reduction: 256 -> 32 via LDS, 32 -> 1 via WMMA ones-trick
    red[tid] = acc;
    __syncthreads();
    if (tid < 32) {                            // wave 0, EXEC all-ones (wave32)
        float p = 0.0f;
#pragma unroll
        for (int k = 0; k < NTHREADS / 32; ++k) p += red[tid + 32 * k];
        const float tot = wave_sum32(p);
        if (tid == 0) partials[bid] = tot;
    }
}

__global__ void __launch_bounds__(NTHREADS)
final_reduce_kernel(const float* __restrict__ partials, float* __restrict__ out) {
    __shared__ float red[NTHREADS];
    float s = 0.0f;
    for (int i = threadIdx.x; i < NBLOCKS; i += NTHREADS) s += partials[i];
    red[threadIdx.x] = s;
    __syncthreads();
    if (threadIdx.x < 32) {
        float p = 0.0f;
#pragma unroll
        for (int k = 0; k < NTHREADS / 32; ++k) p += red[threadIdx.x + 32 * k];
        const float tot = wave_sum32(p);
        if (threadIdx.x == 0)
            out[0] = tot * (1.0f / (16.0f * 3.0f * 512.0f * 512.0f));
    }
}

extern "C" void kernel_launch(void* const* d_in, const int* in_sizes, int n_in,
                              void* d_out, int out_size, void* d_ws, size_t ws_size,
                              hipStream_t stream) {
    (void)in_sizes; (void)n_in; (void)out_size; (void)ws_size;
    const float* img1 = (const float*)d_in[0];
    const float* img2 = (const float*)d_in[1];
    float* out      = (float*)d_out;
    float* partials = (float*)d_ws;   // NBLOCKS floats = 12 KB, fully rewritten each call

    edge_tile_kernel<<<NBLOCKS, NTHREADS, 0, stream>>>(img1, img2, partials);
    final_reduce_kernel<<<1, NTHREADS, 0, stream>>>(partials, out);
}